// KLDRepPointsLoss_6502580486436
// MI455X (gfx1250) — compile-verified
//
#include <hip/hip_runtime.h>

// CDNA5 / gfx1250. wave32. Memory-bound streaming KL-divergence reduction:
//   floor = 1M * 104B / 23.3 TB/s ~= 4.7 us. Compute is negligible.
// Wave-level reduction uses V_WMMA_F32_16X16X4_F32 (A = ones) so the matrix
// unit performs the cross-lane summation.

typedef __attribute__((ext_vector_type(2))) float v2f;
typedef __attribute__((ext_vector_type(8))) float v8f;

// Sum x across all 32 lanes of the wave using one WMMA + 4 xor-shuffles.
// A(16x4) = all ones, B(4x16): lane's x in VGPR0 slot, 0 in VGPR1 slot.
// D[m][n] = sum_k B[k][n]; row 0 of D (lanes 0..15, and duplicated in lanes
// 16..31 as row 8) holds the 16 distinct column sums; xor-reduce masks 1,2,4,8
// sum those 16 columns inside each half-wave -> full 32-lane total in every lane.
__device__ __forceinline__ float wave_sum32(float x) {
    v2f a = {1.0f, 1.0f};   // all-ones A matrix (layout-independent)
    v2f b = {x, 0.0f};      // each lane contributes exactly once via B VGPR0
    v8f c = {};
    v8f d = __builtin_amdgcn_wmma_f32_16x16x4_f32(
        /*neg_a=*/false, a, /*neg_b=*/false, b,
        /*c_mod=*/(short)0, c, /*reuse_a=*/false, /*reuse_b=*/false);
    float cs = d[0];
    cs += __shfl_xor(cs, 1, 32);
    cs += __shfl_xor(cs, 2, 32);
    cs += __shfl_xor(cs, 4, 32);
    cs += __shfl_xor(cs, 8, 32);
    return cs;
}

__global__ void __launch_bounds__(256) kld_partial_kernel(
    const float* __restrict__ pred,   // (n, 9, 2) -> 72 B/sample, 8B-aligned
    const float* __restrict__ targ,   // (n, 4, 2) -> 32 B/sample, 16B-aligned
    float* __restrict__ partials,     // one float per block
    int n) {
    const float INV9  = 1.0f / 9.0f;
    const float INV36 = 1.0f / 36.0f;   // 1/(4*L*L), L=3
    const float GEPS  = 1e-6f;

    float acc = 0.0f;
    int stride = (int)(gridDim.x * blockDim.x);
    for (int i = (int)(blockIdx.x * blockDim.x + threadIdx.x); i < n; i += stride) {
        // ---- pred: mean + covariance of 9 points ----
        const float* p = pred + (size_t)i * 18u;
        float2 pt[9];
        float sx = 0.0f, sy = 0.0f;
#pragma unroll
        for (int k = 0; k < 9; ++k) {
            pt[k] = *(const float2*)(p + 2 * k);   // global_load_b64, coalesced stream
            sx += pt[k].x;
            sy += pt[k].y;
        }
        float mux = sx * INV9, muy = sy * INV9;
        float sxx = 0.0f, sxy = 0.0f, syy = 0.0f;
#pragma unroll
        for (int k = 0; k < 9; ++k) {
            float dx = pt[k].x - mux;
            float dy = pt[k].y - muy;
            sxx += dx * dx;
            sxy += dx * dy;
            syy += dy * dy;
        }
        float pa = sxx * INV9 + GEPS;   // p_var = [[pa, pb],[pb, pd]]
        float pb = sxy * INV9;
        float pd = syy * INV9 + GEPS;
        float p_det = pa * pd - pb * pb;

        // ---- target: oriented box -> rotated diagonal covariance ----
        const float* t = targ + (size_t)i * 8u;
        float4 t01 = *(const float4*)(t);       // (t0x,t0y,t1x,t1y)
        float4 t23 = *(const float4*)(t + 4);   // (t2x,t2y,t3x,t3y)
        float tmux = 0.25f * (t01.x + t01.z + t23.x + t23.z);
        float tmuy = 0.25f * (t01.y + t01.w + t23.y + t23.w);
        float e1x = t01.z - t01.x, e1y = t01.w - t01.y;
        float e2x = t23.x - t01.z, e2y = t23.y - t01.w;
        float w = e1x * e1x + e1y * e1y;
        float h = e2x * e2x + e2y * e2y;
        float rs = 1.0f / sqrtf(w);
        float cth = e1x * rs, sth = e1y * rs;           // R = [[c,-s],[s,c]]
        float dw = w * INV36, dh = h * INV36;
        // t_var = R diag(dw,dh) R^T
        float t00 = cth * cth * dw + sth * sth * dh;
        float t01v = cth * sth * (dw - dh);
        float t11 = sth * sth * dw + cth * cth * dh;
        float t_det = t00 * t11 - t01v * t01v;
        float inv_tdet = 1.0f / t_det;

        // ---- KL divergence terms ----
        float dx = mux - tmux, dy = muy - tmuy;
        float term1 = (t11 * dx * dx - 2.0f * t01v * dx * dy + t00 * dy * dy) * inv_tdet;
        float trace = (t11 * pa - 2.0f * t01v * pb + t00 * pd) * inv_tdet;
        float term2 = trace + logf(t_det / p_det);
        float kld = 0.5f * (term1 + term2) - 1.0f;
        float kl = fmaxf(kld, 1e-6f);
        float loss = 1.0f - 1.0f / (2.0f + sqrtf(kl));
        acc += loss;
    }

    // ---- wave reduce (WMMA) -> block reduce (LDS) -> one partial per block ----
    float wsum = wave_sum32(acc);       // EXEC all-ones here (reconverged)
    __shared__ float lds[8];
    int lane = (int)(threadIdx.x & 31u);
    int wid  = (int)(threadIdx.x >> 5u);
    if (lane == 0) lds[wid] = wsum;
    __syncthreads();
    if (threadIdx.x == 0) {
        float b = 0.0f;
#pragma unroll
        for (int wv = 0; wv < 8; ++wv) b += lds[wv];
        partials[blockIdx.x] = b;
    }
}

__global__ void __launch_bounds__(256) kld_finalize_kernel(
    const float* __restrict__ partials, int nparts,
    float* __restrict__ out, float inv_n) {
    __shared__ float sm[256];
    float s = 0.0f;
    for (int i = (int)threadIdx.x; i < nparts; i += 256) s += partials[i];
    sm[threadIdx.x] = s;
    __syncthreads();
#pragma unroll
    for (int off = 128; off > 0; off >>= 1) {
        if ((int)threadIdx.x < off) sm[threadIdx.x] += sm[threadIdx.x + off];
        __syncthreads();
    }
    if (threadIdx.x == 0) out[0] = sm[0] * inv_n;
}

extern "C" void kernel_launch(void* const* d_in, const int* in_sizes, int n_in,
                              void* d_out, int out_size, void* d_ws, size_t ws_size,
                              hipStream_t stream) {
    const float* pred = (const float*)d_in[0];   // (N, 9, 2) fp32
    const float* targ = (const float*)d_in[1];   // (N, 4, 2) fp32
    float* out = (float*)d_out;                  // scalar fp32
    int n = in_sizes[0] / 18;

    int nblocks = 2048;                          // 2 samples/thread at N = 1M
    int needed = (n + 255) / 256;
    if (nblocks > needed) nblocks = needed;
    int maxp = (int)(ws_size / sizeof(float));
    if (nblocks > maxp) nblocks = (maxp > 0) ? maxp : 1;
    if (nblocks < 1) nblocks = 1;

    float* partials = (float*)d_ws;              // fully rewritten each launch

    kld_partial_kernel<<<nblocks, 256, 0, stream>>>(pred, targ, partials, n);
    kld_finalize_kernel<<<1, 256, 0, stream>>>(partials, nblocks, out,
                                               1.0f / (float)n);
}